// CapsuleParall_61160334295612
// MI455X (gfx1250) — compile-verified
//
#include <hip/hip_runtime.h>
#include <math.h>

typedef __attribute__((ext_vector_type(2))) float v2f;
typedef __attribute__((ext_vector_type(8))) float v8f;

#define IN_F   1152
#define OUT_F  128
#define NUM_C  16
#define BATCH  32
#define EPS_V  1e-5f

// ---------------------------------------------------------------------------
// Phase 1: T[b,n,o] = sum_i u[b,n,i] * (W[n,i,o] * c0[i,o])
// One wave per 16x16 output tile, K-loop of V_WMMA_F32_16X16X4_F32.
// A (16x4 f32):  lanes 0-15 hold M=lane, K={k,k+1}; lanes 16-31 M=lane-16, K={k+2,k+3}
// B (4x16 f32):  lanes 0-15 hold N=lane, K rows {k,k+1}; lanes 16-31 K rows {k+2,k+3}
// D (16x16 f32): VGPR r -> M = r (lanes 0-15) / r+8 (lanes 16-31), N = lane&15
// ---------------------------------------------------------------------------
__global__ __launch_bounds__(32)
void caps_phase1_wmma(const float* __restrict__ u,
                      const float* __restrict__ W,
                      const float* __restrict__ c0,
                      float* __restrict__ T) {
  const int n  = blockIdx.x;        // capsule index
  const int m0 = blockIdx.y * 16;   // batch tile base
  const int n0 = blockIdx.z * 16;   // out_f tile base
  const int L  = threadIdx.x;       // lane 0..31 (wave32)

  const int half = L >> 4;          // 0: lanes 0-15, 1: lanes 16-31
  const int l15  = L & 15;
  const int mrow = m0 + l15;
  const int ncol = n0 + l15;
  const int koff = half * 2;

  const float* Arow = u + ((size_t)mrow * NUM_C + n) * IN_F;
  const float* Wn   = W + (size_t)n * IN_F * OUT_F;

  v8f c = {0.f, 0.f, 0.f, 0.f, 0.f, 0.f, 0.f, 0.f};
  for (int k = 0; k < IN_F; k += 4) {
    const int kr = k + koff;
    v2f a, b;
    a.x = Arow[kr];
    a.y = Arow[kr + 1];
    b.x = Wn[(size_t)kr * OUT_F + ncol]       * c0[(size_t)kr * OUT_F + ncol];
    b.y = Wn[(size_t)(kr + 1) * OUT_F + ncol] * c0[(size_t)(kr + 1) * OUT_F + ncol];
    // 8 args: (neg_a, A, neg_b, B, c_mod, C, reuse_a, reuse_b)
    c = __builtin_amdgcn_wmma_f32_16x16x4_f32(false, a, false, b, (short)0, c,
                                              false, false);
  }

#pragma unroll
  for (int r = 0; r < 8; ++r) {
    const int m = m0 + r + half * 8;
    T[((size_t)m * NUM_C + n) * OUT_F + ncol] = c[r];
  }
}

// ---------------------------------------------------------------------------
// Phase 2: dynamic routing, fully fused per (b,n). 8 waves; each wave owns
// rows i = wave, wave+8, ...; 4 out_f values per lane (o = lane + 32*q).
// Row softmax reduced inside the wave with __shfl_xor; cross-wave s[] merge
// via LDS float atomics (ds_add_f32).
// ---------------------------------------------------------------------------
__global__ __launch_bounds__(256)
void caps_routing(const float* __restrict__ u,
                  const float* __restrict__ W,
                  const float* __restrict__ bias,
                  const int*  __restrict__ routings,
                  const float* __restrict__ T,
                  float* __restrict__ out) {
  const int n    = blockIdx.x;
  const int b    = blockIdx.y;
  const int tid  = threadIdx.x;
  const int lane = tid & 31;
  const int wave = tid >> 5;

  __shared__ float s_u[IN_F];
  __shared__ float s_vsum[OUT_F];
  __shared__ float s_acc[OUT_F];
  __shared__ float s_norm2;

  const float* Wn   = W    + (size_t)n * IN_F * OUT_F;
  const float* urow = u    + ((size_t)b * NUM_C + n) * IN_F;
  const float* bn   = bias + (size_t)n * OUT_F;
  float*       orow = out  + ((size_t)b * NUM_C + n) * OUT_F;
  const int R = routings[0];

  for (int i = tid; i < IN_F; i += 256) s_u[i] = urow[i];

  // ---- v1 = squash(T + bias) ----
  if (tid < OUT_F)
    s_acc[tid] = T[((size_t)b * NUM_C + n) * OUT_F + tid] + bn[tid];
  __syncthreads();
  if (wave == 0) {
    float t = 0.f;
#pragma unroll
    for (int q = 0; q < 4; ++q) { float x = s_acc[lane + 32 * q]; t += x * x; }
#pragma unroll
    for (int off = 16; off > 0; off >>= 1) t += __shfl_xor(t, off);
    if (lane == 0) s_norm2 = t;
  }
  __syncthreads();
  if (tid < OUT_F) {
    const float n2 = s_norm2;
    const float v  = n2 / (1.f + n2) * s_acc[tid] / (sqrtf(n2) + EPS_V);
    s_vsum[tid] = v;
    if (R == 1) orow[tid] = v;
  }
  __syncthreads();
  if (R == 1) return;   // uniform branch across the block

  for (int r = 1; r <= R - 1; ++r) {
    if (tid < OUT_F) s_acc[tid] = 0.f;
    __syncthreads();

    float vs[4];
#pragma unroll
    for (int q = 0; q < 4; ++q) vs[q] = s_vsum[lane + 32 * q];

    float accv[4] = {0.f, 0.f, 0.f, 0.f};
    for (int i = wave; i < IN_F; i += 8) {
      const float  ui   = s_u[i];
      const float* wrow = Wn + (size_t)i * OUT_F;
      float uh[4], p[4];
      float m = -3.402823466e38f;
#pragma unroll
      for (int q = 0; q < 4; ++q) {
        uh[q] = ui * wrow[lane + 32 * q];   // u_hat[i,o]
        p[q]  = uh[q] * vs[q];              // logits b[i,o]
        m = fmaxf(m, p[q]);
      }
#pragma unroll
      for (int off = 16; off > 0; off >>= 1) m = fmaxf(m, __shfl_xor(m, off));
      float Z = 0.f;
#pragma unroll
      for (int q = 0; q < 4; ++q) { p[q] = __expf(p[q] - m); Z += p[q]; }
#pragma unroll
      for (int off = 16; off > 0; off >>= 1) Z += __shfl_xor(Z, off);
      const float invZ = 1.f / Z;
#pragma unroll
      for (int q = 0; q < 4; ++q) accv[q] += uh[q] * (p[q] * invZ);
    }
#pragma unroll
    for (int q = 0; q < 4; ++q) atomicAdd(&s_acc[lane + 32 * q], accv[q]);
    __syncthreads();

    if (tid < OUT_F) s_acc[tid] += bn[tid];
    __syncthreads();
    if (wave == 0) {
      float t = 0.f;
#pragma unroll
      for (int q = 0; q < 4; ++q) { float x = s_acc[lane + 32 * q]; t += x * x; }
#pragma unroll
      for (int off = 16; off > 0; off >>= 1) t += __shfl_xor(t, off);
      if (lane == 0) s_norm2 = t;
    }
    __syncthreads();
    if (tid < OUT_F) {
      const float n2 = s_norm2;
      const float v  = n2 / (1.f + n2) * s_acc[tid] / (sqrtf(n2) + EPS_V);
      if (r < R - 1) s_vsum[tid] += v;
      else           orow[tid]   = v;
    }
    __syncthreads();
  }
}

// ---------------------------------------------------------------------------
extern "C" void kernel_launch(void* const* d_in, const int* in_sizes, int n_in,
                              void* d_out, int out_size, void* d_ws, size_t ws_size,
                              hipStream_t stream) {
  (void)in_sizes; (void)n_in; (void)out_size; (void)ws_size;
  const float* u        = (const float*)d_in[0];  // [B, num, in_f]
  const float* W        = (const float*)d_in[1];  // [1, num, in_f, out_f]
  const float* bias     = (const float*)d_in[2];  // [1, num, out_f]
  const float* c0       = (const float*)d_in[3];  // [1, in_f, out_f]
  const int*   routings = (const int*)d_in[4];    // scalar

  float* T = (float*)d_ws;                        // [B, num, out_f] = 256 KB

  caps_phase1_wmma<<<dim3(NUM_C, BATCH / 16, OUT_F / 16), 32, 0, stream>>>(
      u, W, c0, T);
  caps_routing<<<dim3(NUM_C, BATCH), 256, 0, stream>>>(
      u, W, bias, routings, T, (float*)d_out);
}